// InvariantPointAttention_44375602102613
// MI455X (gfx1250) — compile-verified
//
#include <hip/hip_runtime.h>
#include <hip/hip_bf16.h>
#include <math.h>

// ---------------------------------------------------------------------------
// Invariant Point Attention for MI455X (gfx1250), fp32 end-to-end.
// All GEMM-shaped contractions use V_WMMA_F32_16X16X4_F32 (wave32 WMMA).
// Problem constants from the reference:
//   B=1, N=768, C_S=384, C_Z=128, H=12, D=32, PQK=4, PV=8
//
// Roofline: ~7.3 GFLOP total but HBM traffic dominated by the fp32 z tensor
// (302 MB, read exactly twice: z@Wb and a@z) -> memory-bound at 23.3 TB/s.
// fp32 WMMA matches reference numerics; inner loops are branch-free vmem.
// Multi-column tiling (NT) keeps A-panel re-reads off the memory system.
// ---------------------------------------------------------------------------

#define NTOK   768
#define CS     384
#define CZ     128
#define NH     12
#define HD     32
#define PQK    4
#define PV     8
#define HPQ    (NH * PQK)              // 48
#define HPKV   (NH * (PQK + PV))       // 144
#define OUTCAT (NH * (CZ + HD + PV*4)) // 2304

typedef __attribute__((ext_vector_type(2))) float v2f;
typedef __attribute__((ext_vector_type(8))) float v8f;

// ---------------------------------------------------------------------------
// Generic batched fp32 WMMA GEMM:  C[b][m][n] = sum_k A[b][m][k]*B[b][k][n] (+bias[n])
//   A element: A + b*aBS + m*lda + k      (row-major M x K)
//   B element: B + b*bBS + k*ldb + n      (row-major K x N view)
//   C element: C + b*cBS + m*ldc + n
// One wave computes a 16 x (16*NT) strip: each A fragment is loaded once and
// applied to NT B fragments (A reuse x NT). K % 16 == 0 at every call site.
// Ragged M/N: OOB lanes clamp their row/col index (loads stay valid and
// branch-free; bogus products land only in elements whose store is guarded).
// EXEC is all-ones at every v_wmma as the ISA requires.
// ---------------------------------------------------------------------------
template <int NT>
__global__ __launch_bounds__(32)
void wmma_gemm_f32(const float* __restrict__ A, long long lda, long long aBS,
                   const float* __restrict__ B, long long ldb, long long bBS,
                   const float* __restrict__ bias,
                   float* __restrict__ C, long long ldc, long long cBS,
                   int M, int N, int K)
{
    const int bz = blockIdx.z;
    A += (size_t)bz * aBS;
    B += (size_t)bz * bBS;
    C += (size_t)bz * cBS;

    const int i0   = blockIdx.y * 16;
    const int j0   = blockIdx.x * (16 * NT);
    const int lane = threadIdx.x & 31;
    const int half = lane >> 4;     // A/B fragment K-pair select (ISA layout)
    const int lr   = lane & 15;

    const int arow = min(i0 + lr, M - 1);   // clamped: branch-free OOB
    const float* Arow = A + (size_t)arow * lda + 2 * half;  // K-contiguous pair

    const float* Bcol[NT];
    #pragma unroll
    for (int t = 0; t < NT; ++t) {
        const int bcol = min(j0 + t * 16 + lr, N - 1);
        Bcol[t] = B + (size_t)(2 * half) * ldb + bcol;
    }

    v8f acc[NT] = {};
    for (int k0 = 0; k0 < K; k0 += 16) {
        #pragma unroll
        for (int u = 0; u < 4; ++u) {
            const int ka = k0 + 4 * u;
            const v2f a = *(const v2f*)(Arow + ka);          // global_load_b64
            #pragma unroll
            for (int t = 0; t < NT; ++t) {
                v2f b;
                b.x = Bcol[t][(size_t)ka * ldb];
                b.y = Bcol[t][(size_t)(ka + 1) * ldb];
                acc[t] = __builtin_amdgcn_wmma_f32_16x16x4_f32(
                             false, a, false, b, (short)0, acc[t], false, false);
            }
        }
    }

    // C/D layout: VGPR r holds M = i0+r (lanes 0-15) and M = i0+r+8 (lanes 16-31)
    #pragma unroll
    for (int t = 0; t < NT; ++t) {
        const int n = j0 + t * 16 + lr;
        if (n < N) {                                  // loop-invariant per lane
            const float bv = bias ? bias[n] : 0.0f;   // hoisted bias load
            #pragma unroll
            for (int r = 0; r < 8; ++r) {
                const int m = i0 + r + 8 * half;
                if (m < M) C[(size_t)m * ldc + n] = acc[t][r] + bv;
            }
        }
    }
}

// ---------------------------------------------------------------------------
// Rotate raw projected points: x[(i, c*HP + j)] -> out[i*HP*3 + j*3 + c]
//   out = rot[i] @ p + trans[i]
// ---------------------------------------------------------------------------
__global__ void rotate_points(const float* __restrict__ x,
                              const float* __restrict__ rot,
                              const float* __restrict__ trans,
                              float* __restrict__ out, int hp)
{
    const int idx = blockIdx.x * blockDim.x + threadIdx.x;
    if (idx >= NTOK * hp) return;
    const int i = idx / hp;
    const int j = idx - i * hp;
    const float px = x[(size_t)i * (3 * hp) + 0 * hp + j];
    const float py = x[(size_t)i * (3 * hp) + 1 * hp + j];
    const float pz = x[(size_t)i * (3 * hp) + 2 * hp + j];
    const float* R = rot + (size_t)i * 9;
    float* o = out + (size_t)idx * 3;
    o[0] = R[0] * px + R[1] * py + R[2] * pz + trans[i * 3 + 0];
    o[1] = R[3] * px + R[4] * py + R[5] * pz + trans[i * 3 + 1];
    o[2] = R[6] * px + R[7] * py + R[8] * pz + trans[i * 3 + 2];
}

// ---------------------------------------------------------------------------
// Per-(token, head) squared norm over the 12 floats (PQK*3) of the query/key
// points: out[i*NH + h] = sum_t pts[i*ld + h*hs + t]^2, t in [0,12)
// ---------------------------------------------------------------------------
__global__ void point_sqnorm(const float* __restrict__ pts, int ld, int hs,
                             float* __restrict__ out)
{
    const int idx = blockIdx.x * blockDim.x + threadIdx.x;
    if (idx >= NTOK * NH) return;
    const int i = idx / NH;
    const int h = idx - i * NH;
    const float* p = pts + (size_t)i * ld + h * hs;
    float s = 0.0f;
    #pragma unroll
    for (int t = 0; t < 12; ++t) s += p[t] * p[t];
    out[idx] = s;
}

// ---------------------------------------------------------------------------
// Fused attention logits: per (head h, 16x16 (i,j) tile) one wave.
//   logit = (q.k)/sqrt(D) + zbias - 0.5*pt_scale*softplus(w_h)*(qn+kn-2*cross)
//   logit *= sqrt(1/3);  logit += INF*(mask_i*mask_j - 1)
// q.k is fp32 WMMA over K=D=32 (8 wmma); cross over K=PQK*3=12 (3 wmma).
// Both operands are K-contiguous within a token row -> b64 loads.
// ---------------------------------------------------------------------------
__global__ __launch_bounds__(32)
void ipa_scores(const float* __restrict__ q,     // [N][H*D]
                const float* __restrict__ kv,    // [N][H*2D], k at h*64, v at h*64+32
                const float* __restrict__ qpts,  // [N][HPQ*3]
                const float* __restrict__ kvpts, // [N][HPKV*3]
                const float* __restrict__ qn,    // [N][H]
                const float* __restrict__ kn,    // [N][H]
                const float* __restrict__ zb,    // [N*N][H]
                const float* __restrict__ mask,  // [N]
                const float* __restrict__ hwts,  // [H]
                float* __restrict__ attn)        // [H][N][N]
{
    const int h    = blockIdx.z;
    const int i0   = blockIdx.y * 16;
    const int j0   = blockIdx.x * 16;
    const int lane = threadIdx.x & 31;
    const int half = lane >> 4;
    const int lr   = lane & 15;

    v8f qk = {};
    v8f cr = {};

    // q . k over D=32
    const float* qrow = q  + (size_t)(i0 + lr) * CS       + h * HD       + 2 * half;
    const float* krow = kv + (size_t)(j0 + lr) * (2 * CS) + h * (2 * HD) + 2 * half;
    #pragma unroll
    for (int k0 = 0; k0 < HD; k0 += 4) {
        const v2f a = *(const v2f*)(qrow + k0);
        const v2f b = *(const v2f*)(krow + k0);
        qk = __builtin_amdgcn_wmma_f32_16x16x4_f32(
                 false, a, false, b, (short)0, qk, false, false);
    }

    // q_pts . k_pts over PQK*3 = 12
    const float* qprow = qpts  + (size_t)(i0 + lr) * (HPQ * 3)  + h * (PQK * 3)        + 2 * half;
    const float* kprow = kvpts + (size_t)(j0 + lr) * (HPKV * 3) + h * ((PQK + PV) * 3) + 2 * half;
    #pragma unroll
    for (int k0 = 0; k0 < 12; k0 += 4) {
        const v2f a = *(const v2f*)(qprow + k0);
        const v2f b = *(const v2f*)(kprow + k0);
        cr = __builtin_amdgcn_wmma_f32_16x16x4_f32(
                 false, a, false, b, (short)0, cr, false, false);
    }

    const float hw       = logf(1.0f + expf(hwts[h]));          // softplus
    const float inv_sd   = 0.17677669529663687f;                // 1/sqrt(32)
    const float pt_scale = 0.23570226039551584f;                // sqrt(1/(PQK*9/2))
    const float rsqrt3   = 0.57735026918962576f;                // sqrt(1/3)
    const float wcoef    = -0.5f * pt_scale * hw;

    const int   j   = j0 + lr;
    const float knv = kn[(size_t)j * NH + h];
    const float mj  = mask[j];

    #pragma unroll
    for (int r = 0; r < 8; ++r) {
        const int i = i0 + r + 8 * half;
        const float qnv = qn[(size_t)i * NH + h];
        float logit = qk[r] * inv_sd
                    + zb[((size_t)i * NTOK + j) * NH + h]
                    + wcoef * (qnv + knv - 2.0f * cr[r]);
        logit *= rsqrt3;
        logit += 100000.0f * (mask[i] * mj - 1.0f);
        attn[(size_t)h * NTOK * NTOK + (size_t)i * NTOK + j] = logit;
    }
}

// ---------------------------------------------------------------------------
// Row softmax over ncols=768, one block of 256 threads per row, in place.
// ---------------------------------------------------------------------------
__global__ __launch_bounds__(256)
void softmax_rows(float* __restrict__ p, int ncols)
{
    __shared__ float red[256];
    float* row = p + (size_t)blockIdx.x * ncols;
    const int t = threadIdx.x;

    float m = -3.0e38f;
    for (int c = t; c < ncols; c += 256) m = fmaxf(m, row[c]);
    red[t] = m; __syncthreads();
    for (int s = 128; s > 0; s >>= 1) {
        if (t < s) red[t] = fmaxf(red[t], red[t + s]);
        __syncthreads();
    }
    m = red[0]; __syncthreads();

    float sum = 0.0f;
    for (int c = t; c < ncols; c += 256) {
        float e = expf(row[c] - m);
        row[c] = e;
        sum += e;
    }
    red[t] = sum; __syncthreads();
    for (int s = 128; s > 0; s >>= 1) {
        if (t < s) red[t] += red[t + s];
        __syncthreads();
    }
    const float inv = 1.0f / red[0];
    for (int c = t; c < ncols; c += 256) row[c] *= inv;
}

// ---------------------------------------------------------------------------
// Inverse-rotate attended points, compute norms, scatter into concat buffer X.
//   X[i] = [ o(384) | o_pt.x(96) | o_pt.y(96) | o_pt.z(96) | norm(96) | o_pair(1536) ]
// op layout: [i][h*24 + p*3 + c]   (h in [0,12), p in [0,8))
// ---------------------------------------------------------------------------
__global__ void ipa_finish_points(const float* __restrict__ op,
                                  const float* __restrict__ rot,
                                  const float* __restrict__ trans,
                                  float* __restrict__ X)
{
    const int idx = blockIdx.x * blockDim.x + threadIdx.x;
    if (idx >= NTOK * NH * PV) return;
    const int i  = idx / (NH * PV);
    const int hp = idx - i * (NH * PV);          // h*PV + p
    const int h  = hp >> 3;
    const int pp = hp & 7;

    const float* v = op + (size_t)i * (NH * PV * 3) + h * (PV * 3) + pp * 3;
    const float* R = rot + (size_t)i * 9;
    const float vx = v[0] - trans[i * 3 + 0];
    const float vy = v[1] - trans[i * 3 + 1];
    const float vz = v[2] - trans[i * 3 + 2];
    // R^T * v
    const float ox = R[0] * vx + R[3] * vy + R[6] * vz;
    const float oy = R[1] * vx + R[4] * vy + R[7] * vz;
    const float oz = R[2] * vx + R[5] * vy + R[8] * vz;
    const float nm = sqrtf(ox * ox + oy * oy + oz * oz + 1e-8f);

    float* xr = X + (size_t)i * OUTCAT + NH * HD;  // skip o block (384)
    xr[hp]       = ox;
    xr[96 + hp]  = oy;
    xr[192 + hp] = oz;
    xr[288 + hp] = nm;
}

// ---------------------------------------------------------------------------
extern "C" void kernel_launch(void* const* d_in, const int* in_sizes, int n_in,
                              void* d_out, int out_size, void* d_ws, size_t ws_size,
                              hipStream_t stream)
{
    const float* s     = (const float*)d_in[0];
    const float* z     = (const float*)d_in[1];
    const float* rot   = (const float*)d_in[2];
    const float* trans = (const float*)d_in[3];
    const float* mask  = (const float*)d_in[4];
    const float* Wq    = (const float*)d_in[5];
    const float* bq    = (const float*)d_in[6];
    const float* Wkv   = (const float*)d_in[7];
    const float* bkv   = (const float*)d_in[8];
    const float* Wqp   = (const float*)d_in[9];
    const float* bqp   = (const float*)d_in[10];
    const float* Wkvp  = (const float*)d_in[11];
    const float* bkvp  = (const float*)d_in[12];
    const float* Wb    = (const float*)d_in[13];
    const float* bb    = (const float*)d_in[14];
    const float* hwts  = (const float*)d_in[15];
    const float* Wout  = (const float*)d_in[16];
    const float* bout  = (const float*)d_in[17];
    float* out = (float*)d_out;

    // ---- workspace layout (floats), total ~72 MB (fits global L2) ----
    float* W      = (float*)d_ws;
    float* q      = W;                        // 768*384
    float* kv     = q      + NTOK * CS;       // 768*768
    float* qpraw  = kv     + NTOK * 2 * CS;   // 768*144
    float* kvpraw = qpraw  + NTOK * HPQ * 3;  // 768*432
    float* qpts   = kvpraw + NTOK * HPKV * 3; // 768*144
    float* kvpts  = qpts   + NTOK * HPQ * 3;  // 768*432
    float* qn     = kvpts  + NTOK * HPKV * 3; // 768*12
    float* kn     = qn     + NTOK * NH;       // 768*12
    float* zb     = kn     + NTOK * NH;       // 768*768*12
    float* attn   = zb     + (size_t)NTOK * NTOK * NH;  // 12*768*768
    float* op     = attn   + (size_t)NH * NTOK * NTOK;  // 768*288
    float* X      = op     + NTOK * NH * PV * 3;        // 768*2304

    auto gemm = [&](int NT,
                    const float* A, long long lda, long long aBS,
                    const float* B, long long ldb, long long bBS,
                    const float* bias, float* C, long long ldc, long long cBS,
                    int M, int N, int K, int batches) {
        dim3 g((unsigned)((N + 16 * NT - 1) / (16 * NT)),
               (unsigned)((M + 15) / 16), (unsigned)batches);
        switch (NT) {
        case 2:
            wmma_gemm_f32<2><<<g, dim3(32), 0, stream>>>(A, lda, aBS, B, ldb, bBS,
                                                         bias, C, ldc, cBS, M, N, K);
            break;
        case 4:
            wmma_gemm_f32<4><<<g, dim3(32), 0, stream>>>(A, lda, aBS, B, ldb, bBS,
                                                         bias, C, ldc, cBS, M, N, K);
            break;
        case 8:
            wmma_gemm_f32<8><<<g, dim3(32), 0, stream>>>(A, lda, aBS, B, ldb, bBS,
                                                         bias, C, ldc, cBS, M, N, K);
            break;
        default:
            wmma_gemm_f32<1><<<g, dim3(32), 0, stream>>>(A, lda, aBS, B, ldb, bBS,
                                                         bias, C, ldc, cBS, M, N, K);
            break;
        }
    };

    // 1) Projections of s
    gemm(4, s, CS, 0, Wq,   CS,       0, bq,   q,      CS,       0, NTOK, CS,       CS, 1);
    gemm(4, s, CS, 0, Wkv,  2 * CS,   0, bkv,  kv,     2 * CS,   0, NTOK, 2 * CS,   CS, 1);
    gemm(1, s, CS, 0, Wqp,  HPQ * 3,  0, bqp,  qpraw,  HPQ * 3,  0, NTOK, HPQ * 3,  CS, 1);
    gemm(1, s, CS, 0, Wkvp, HPKV * 3, 0, bkvp, kvpraw, HPKV * 3, 0, NTOK, HPKV * 3, CS, 1);

    // 2) Rotate points into world frame
    {
        int tot = NTOK * HPQ;
        rotate_points<<<(tot + 255) / 256, 256, 0, stream>>>(qpraw, rot, trans, qpts, HPQ);
        tot = NTOK * HPKV;
        rotate_points<<<(tot + 255) / 256, 256, 0, stream>>>(kvpraw, rot, trans, kvpts, HPKV);
    }

    // 3) Point squared norms (query pts; key pts = first 12 floats per head)
    {
        const int tot = NTOK * NH;
        point_sqnorm<<<(tot + 255) / 256, 256, 0, stream>>>(qpts,  HPQ * 3,  PQK * 3,        qn);
        point_sqnorm<<<(tot + 255) / 256, 256, 0, stream>>>(kvpts, HPKV * 3, (PQK + PV) * 3, kn);
    }

    // 4) z-bias: zb[(i*N+j)][h] = z[i][j][:] . Wb[:,h] + bb[h]   (reads z once)
    gemm(1, z, CZ, 0, Wb, NH, 0, bb, zb, NH, 0, NTOK * NTOK, NH, CZ, 1);

    // 5) Fused logits (WMMA q.k + point term), per head
    {
        dim3 g(NTOK / 16, NTOK / 16, NH);
        ipa_scores<<<g, dim3(32), 0, stream>>>(q, kv, qpts, kvpts, qn, kn, zb,
                                               mask, hwts, attn);
    }

    // 6) Softmax over j, in place
    softmax_rows<<<NH * NTOK, 256, 0, stream>>>(attn, NTOK);

    // 7) o = a @ v  -> X columns [h*32, h*32+32); batch over heads
    gemm(2, attn, NTOK, (long long)NTOK * NTOK,
         kv + HD, 2 * CS, 2 * HD,
         nullptr, X, OUTCAT, HD,
         NTOK, HD, NTOK, NH);

    // 8) o_pt(pre-rot) = a @ v_pts  -> op[i][h*24 + n]; batch over heads
    gemm(1, attn, NTOK, (long long)NTOK * NTOK,
         kvpts + PQK * 3, HPKV * 3, (PQK + PV) * 3,
         nullptr, op, NH * PV * 3, PV * 3,
         NTOK, PV * 3, NTOK, NH);

    // 9) o_pair[i][h][c] = sum_j a[h,i,j] * z[i,j,c]; batch over i (reads z once)
    //    A[m=h][k=j] = attn[h*N*N + i*N + j];  C -> X cols [768, 2304)
    //    NT=8 covers all 128 z-channels in one wave: attn slice read once.
    gemm(8, attn, (long long)NTOK * NTOK, NTOK,
         z, CZ, (long long)NTOK * CZ,
         nullptr, X + NH * HD + 4 * NH * PV /*=768*/, CZ, OUTCAT,
         NH, CZ, NTOK, NTOK);

    // 10) Inverse-rotate points, norms -> X columns [384, 768)
    {
        const int tot = NTOK * NH * PV;
        ipa_finish_points<<<(tot + 255) / 256, 256, 0, stream>>>(op, rot, trans, X);
    }

    // 11) Final projection: out = X @ Wout + bout (NT=8: X panel read 3x not 24x)
    gemm(8, X, OUTCAT, 0, Wout, CS, 0, bout, out, CS, 0, NTOK, CS, OUTCAT, 1);
}